// UntraceableModule3_9947144258008
// MI455X (gfx1250) — compile-verified
//
#include <hip/hip_runtime.h>
#include <hip/hip_bf16.h>

// CDNA5 / gfx1250: wave32, V_WMMA_F32_16X16X4_F32 as a streaming reducer.
// A(16x4)=ones, B(4x16)=data  =>  D[m][n] = colsum(B)[n] + C[m][n].
// Summation is permutation invariant, so any 64 loaded f32s per B operand are
// valid; reducing row 0 of D over its 16 lanes yields the chunk sum in f32.

typedef __attribute__((ext_vector_type(2))) float v2f;
typedef __attribute__((ext_vector_type(8))) float v8f;

#define RED_BLOCKS   1024
#define RED_THREADS  256   // 8 waves/block -> 8192 waves
#define ADD_BLOCKS   2048
#define ADD_THREADS  256
#define CHUNK        256   // floats per wave per iteration (2 x float4 / lane)

// ---------------------------------------------------------------------------
// Stage 1: per-wave partial sums. Hot loop: 2x global_load_b128 + 4x wmma.
// ---------------------------------------------------------------------------
__global__ __launch_bounds__(RED_THREADS)
void reduce_partials_wmma(const float* __restrict__ x, long long n,
                          float* __restrict__ partials) {
    const int lane         = threadIdx.x & 31;
    const int waveInBlock  = threadIdx.x >> 5;
    const int wavesPerBlk  = blockDim.x >> 5;
    const long long waveId = (long long)blockIdx.x * wavesPerBlk + waveInBlock;
    const long long nWaves = (long long)gridDim.x * wavesPerBlk;

    // A = all-ones 16x4 f32: every element 1.0, layout-independent.
    v2f a; a.x = 1.0f; a.y = 1.0f;

    // Four independent accumulator chains.
    v8f acc0 = {}, acc1 = {}, acc2 = {}, acc3 = {};

    // ---- main loop: fully in-bounds chunks only (no divergence) ----
    const long long fullChunks = n / CHUNK;
    for (long long c = waveId; c < fullChunks; c += nWaves) {
        const float4* p = reinterpret_cast<const float4*>(x + c * CHUNK) + lane;
        const float4 va = p[0];    // bytes [lane*16,        +16)
        const float4 vb = p[32];   // bytes [512 + lane*16,  +16)

        v2f b0, b1, b2, b3;
        b0.x = va.x; b0.y = va.y;
        b1.x = va.z; b1.y = va.w;
        b2.x = vb.x; b2.y = vb.y;
        b3.x = vb.z; b3.y = vb.w;

        acc0 = __builtin_amdgcn_wmma_f32_16x16x4_f32(false, a, false, b0,
                                                     (short)0, acc0, false, false);
        acc1 = __builtin_amdgcn_wmma_f32_16x16x4_f32(false, a, false, b1,
                                                     (short)0, acc1, false, false);
        acc2 = __builtin_amdgcn_wmma_f32_16x16x4_f32(false, a, false, b2,
                                                     (short)0, acc2, false, false);
        acc3 = __builtin_amdgcn_wmma_f32_16x16x4_f32(false, a, false, b3,
                                                     (short)0, acc3, false, false);
    }

    // ---- tail (< CHUNK elements), handled once by wave 0 only ----
    if (waveId == 0) {
        const long long tailStart = fullChunks * CHUNK;
        if (tailStart < n) {
            const long long b = tailStart + (long long)lane * 4;
            const long long h = b + 128;
            v2f b0, b1, b2, b3;
            b0.x = (b + 0 < n) ? x[b + 0] : 0.0f;
            b0.y = (b + 1 < n) ? x[b + 1] : 0.0f;
            b1.x = (b + 2 < n) ? x[b + 2] : 0.0f;
            b1.y = (b + 3 < n) ? x[b + 3] : 0.0f;
            b2.x = (h + 0 < n) ? x[h + 0] : 0.0f;
            b2.y = (h + 1 < n) ? x[h + 1] : 0.0f;
            b3.x = (h + 2 < n) ? x[h + 2] : 0.0f;
            b3.y = (h + 3 < n) ? x[h + 3] : 0.0f;
            acc0 = __builtin_amdgcn_wmma_f32_16x16x4_f32(false, a, false, b0,
                                                         (short)0, acc0, false, false);
            acc1 = __builtin_amdgcn_wmma_f32_16x16x4_f32(false, a, false, b1,
                                                         (short)0, acc1, false, false);
            acc2 = __builtin_amdgcn_wmma_f32_16x16x4_f32(false, a, false, b2,
                                                         (short)0, acc2, false, false);
            acc3 = __builtin_amdgcn_wmma_f32_16x16x4_f32(false, a, false, b3,
                                                         (short)0, acc3, false, false);
        }
    }

    // Row 0 of D: lanes 0..15 hold column sums (lanes 16..31 are the
    // replicated M=8 row). Reduce the 16-lane segment.
    float v = (acc0[0] + acc1[0]) + (acc2[0] + acc3[0]);
    #pragma unroll
    for (int off = 8; off >= 1; off >>= 1)
        v += __shfl_xor(v, off, 16);

    if (lane == 0)
        partials[waveId] = v;
}

// ---------------------------------------------------------------------------
// Stage 2: single-block deterministic reduction (double accumulation +
// fixed-order tree), then the reference's f32 scalar math.
// ---------------------------------------------------------------------------
__global__ __launch_bounds__(256)
void finalize_total(const float* __restrict__ partials, int numPartials,
                    float* __restrict__ totalOut) {
    __shared__ double sm[256];
    const int t = threadIdx.x;

    double s = 0.0;
    for (int i = t; i < numPartials; i += 256)
        s += (double)partials[i];
    sm[t] = s;
    __syncthreads();

    #pragma unroll
    for (int stride = 128; stride > 0; stride >>= 1) {
        if (t < stride) sm[t] += sm[t + stride];
        __syncthreads();
    }

    if (t == 0) {
        const float sf    = (float)sm[0];     // reference sums in f32
        const float nn    = truncf(sf);       // jnp.trunc
        const float total = (nn > 1.0f) ? nn * (nn - 1.0f) * 0.5f : 0.0f;
        *totalOut = total;
    }
}

// ---------------------------------------------------------------------------
// Stage 3: out[i] = x[i] + total, float4-vectorized grid-stride.
// ---------------------------------------------------------------------------
__global__ __launch_bounds__(ADD_THREADS)
void add_total(const float* __restrict__ x, float* __restrict__ out,
               long long n, const float* __restrict__ totalPtr) {
    const float total = *totalPtr;   // uniform scalar from d_ws

    const long long tid    = (long long)blockIdx.x * blockDim.x + threadIdx.x;
    const long long stride = (long long)gridDim.x * blockDim.x;
    const long long n4     = n >> 2;

    const float4* __restrict__ x4 = reinterpret_cast<const float4*>(x);
    float4* __restrict__ o4       = reinterpret_cast<float4*>(out);

    for (long long i = tid; i < n4; i += stride) {
        float4 v = x4[i];
        v.x += total; v.y += total; v.z += total; v.w += total;
        o4[i] = v;
    }
    for (long long i = (n4 << 2) + tid; i < n; i += stride)
        out[i] = x[i] + total;
}

// ---------------------------------------------------------------------------
extern "C" void kernel_launch(void* const* d_in, const int* in_sizes, int n_in,
                              void* d_out, int out_size, void* d_ws, size_t ws_size,
                              hipStream_t stream) {
    const float* x = (const float*)d_in[0];
    float* out     = (float*)d_out;
    const long long n = (long long)in_sizes[0];

    // Workspace layout: [numWaves partial floats][1 total float].
    int redBlocks = RED_BLOCKS;
    long long maxWaves = (long long)(ws_size / sizeof(float)) - 1;
    if (maxWaves < 1) maxWaves = 1;
    while ((long long)redBlocks * (RED_THREADS / 32) > maxWaves && redBlocks > 1)
        redBlocks >>= 1;
    const int numWaves = redBlocks * (RED_THREADS / 32);

    float* partials = (float*)d_ws;
    float* totalPtr = partials + numWaves;

    reduce_partials_wmma<<<redBlocks, RED_THREADS, 0, stream>>>(x, n, partials);
    finalize_total<<<1, 256, 0, stream>>>(partials, numWaves, totalPtr);
    add_total<<<ADD_BLOCKS, ADD_THREADS, 0, stream>>>(x, out, n, totalPtr);
}